// GAT_17162689314906
// MI455X (gfx1250) — compile-verified
//
#include <hip/hip_runtime.h>
#include <math.h>

// ---------------------------------------------------------------------------
// 2-layer GAT (PyG GATConv semantics) for MI455X / gfx1250.
//  - Layer-1 projection x@W1 (50000x256 * 256x256): blocked WMMA fp32 kernel.
//    Block = 16 rows x 256 cols, 8 waves, 2 accumulators/wave. A-tile staged
//    in LDS (stride-76 padding -> conflict-free ds_load_b64 of A fragments),
//    W1 read from L2 (256KB, resident). V_WMMA_F32_16X16X4_F32 throughout.
//  - Layer-2 projection h@W2 (256->16): generic one-tile-per-wave WMMA kernel.
//  - Edge segment-softmax + aggregation via float atomics (gathers L2-fed:
//    xp is 51MB << 192MB L2).
// ---------------------------------------------------------------------------

typedef __attribute__((ext_vector_type(2))) float v2f;
typedef __attribute__((ext_vector_type(8))) float v8f;

#define NEG_SLOPE 0.2f
#define EPSV 1e-16f
#define LDSTR 76   // LDS row stride (floats): 16B-aligned rows, conflict-free

// ---- monotone float<->uint encoding so atomicMax(uint) == float max --------
__device__ __forceinline__ unsigned int encOrd(float f) {
  unsigned int u = __float_as_uint(f);
  return (u & 0x80000000u) ? ~u : (u | 0x80000000u);
}
__device__ __forceinline__ float decOrd(unsigned int e) {
  unsigned int u = (e & 0x80000000u) ? (e & 0x7fffffffu) : ~e;
  return __uint_as_float(u);
}

// ---- generic zero fill -----------------------------------------------------
__global__ void fill_zero(float* __restrict__ p, size_t n) {
  size_t i = (size_t)blockIdx.x * blockDim.x + threadIdx.x;
  if (i < n) p[i] = 0.0f;
}

// ---- layer-1 GEMM: C[M,256] = A[M,256] * B[256,256], one 16-row strip/block.
// 8 waves; wave w covers cols [32w, 32w+32) with two 16x16 WMMA accumulators.
__global__ __launch_bounds__(256)
void gemm1_f32_wmma(const float* __restrict__ A, const float* __restrict__ B,
                    float* __restrict__ C, int M) {
  __shared__ float lda[16 * LDSTR];

  const int tM   = blockIdx.x << 4;
  const int w    = threadIdx.x >> 5;     // wave 0..7
  const int lane = threadIdx.x & 31;
  const int m    = lane & 15;
  const int half = lane >> 4;
  const int tN0  = w << 5;               // this wave's first col tile

  const float* __restrict__ b0 = B + tN0 + m;
  const float* __restrict__ b1 = B + tN0 + 16 + m;

  // cooperative-stage indices: thread t loads one float4 of the A strip
  const int srow = threadIdx.x >> 4;           // 0..15
  const int sc4  = (threadIdx.x & 15) << 2;    // 0,4,...,60
  const float* __restrict__ arow = A + (size_t)(tM + srow) * 256 + sc4;

  v8f acc0 = {};
  v8f acc1 = {};

  for (int kc = 0; kc < 256; kc += 64) {
    // stage A[16][64] chunk into LDS (float4 per thread, 16B-aligned rows)
    const float4 av = *(const float4*)(arow + kc);
    *(float4*)&lda[srow * LDSTR + sc4] = av;
    __syncthreads();

#pragma unroll
    for (int k = 0; k < 64; k += 4) {
      const int ka = k + half * 2;
      const v2f a = *(const v2f*)&lda[m * LDSTR + ka];   // ds_load_b64
      const size_t koff = (size_t)(kc + ka) * 256;
      v2f bA, bB;
      bA.x = b0[koff];       bA.y = b0[koff + 256];
      bB.x = b1[koff];       bB.y = b1[koff + 256];
      acc0 = __builtin_amdgcn_wmma_f32_16x16x4_f32(false, a, false, bA,
                                                   (short)0, acc0, false, false);
      acc1 = __builtin_amdgcn_wmma_f32_16x16x4_f32(false, a, false, bB,
                                                   (short)0, acc1, false, false);
    }
    __syncthreads();
  }

#pragma unroll
  for (int r = 0; r < 8; ++r) {
    const size_t row = (size_t)(tM + r + half * 8) * 256;
    C[row + tN0 + m]      = acc0[r];
    C[row + tN0 + 16 + m] = acc1[r];
  }
}

// ---- generic fp32 WMMA GEMM (one 16x16 tile per wave) for layer 2 ----------
__global__ __launch_bounds__(256)
void gemm_f32_wmma(const float* __restrict__ A, const float* __restrict__ B,
                   float* __restrict__ C, int M, int Nc, int K) {
  const int lane = threadIdx.x & 31;
  const int wave = blockIdx.x * (blockDim.x >> 5) + (threadIdx.x >> 5);
  const int tilesN = Nc >> 4;
  const int tiles  = (M >> 4) * tilesN;
  if (wave >= tiles) return;               // wave-uniform: EXEC stays all-ones
  const int tM   = (wave / tilesN) << 4;
  const int tN   = (wave % tilesN) << 4;
  const int m    = lane & 15;
  const int half = lane >> 4;

  const float* __restrict__ arow = A + (size_t)(tM + m) * K;
  const float* __restrict__ bcol = B + tN + m;

  v8f acc = {};
  for (int k = 0; k < K; k += 4) {
    const int ka = k + half * 2;
    v2f a, b;
    a.x = arow[ka];
    a.y = arow[ka + 1];
    b.x = bcol[(size_t)ka * Nc];
    b.y = bcol[(size_t)(ka + 1) * Nc];
    acc = __builtin_amdgcn_wmma_f32_16x16x4_f32(false, a, false, b,
                                                (short)0, acc, false, false);
  }
#pragma unroll
  for (int r = 0; r < 8; ++r)
    C[(size_t)(tM + r + half * 8) * Nc + tN + m] = acc[r];
}

// ---- per-(node,head) attention logits: al = <xp[n,h,:], a[h,:]> ------------
__global__ void attn_logits(const float* __restrict__ xp,
                            const float* __restrict__ a_src,
                            const float* __restrict__ a_dst,
                            float* __restrict__ als, float* __restrict__ ald,
                            int Nn, int H, int C) {
  int idx = blockIdx.x * blockDim.x + threadIdx.x;
  if (idx >= Nn * H) return;
  const int n = idx / H, h = idx % H;
  const float* v  = xp + (size_t)n * H * C + (size_t)h * C;
  const float* as = a_src + h * C;
  const float* ad = a_dst + h * C;
  float s0 = 0.f, s1 = 0.f;
  for (int c = 0; c < C; ++c) { s0 += v[c] * as[c]; s1 += v[c] * ad[c]; }
  als[idx] = s0;
  ald[idx] = s1;
}

// ---- edge helpers ----------------------------------------------------------
__device__ __forceinline__ void edge_sd(const int* __restrict__ ei, int E,
                                        int j, int& s, int& d) {
  if (j < E) { s = ei[j]; d = ei[E + j]; }
  else       { s = j - E; d = j - E; }     // self-loops appended after E edges
}
__device__ __forceinline__ float leaky(float e) {
  return e > 0.f ? e : e * NEG_SLOPE;
}

// ---- pass 1: segment max over destination ----------------------------------
__global__ void edge_seg_max(const int* __restrict__ ei, int E, int Nn, int H,
                             const float* __restrict__ als,
                             const float* __restrict__ ald,
                             unsigned int* __restrict__ menc) {
  int idx = blockIdx.x * blockDim.x + threadIdx.x;
  if (idx >= (E + Nn) * H) return;
  const int j = idx / H, h = idx % H;
  int s, d; edge_sd(ei, E, j, s, d);
  const float e = leaky(als[s * H + h] + ald[d * H + h]);
  atomicMax(&menc[d * H + h], encOrd(e));
}

// ---- pass 2: ex = exp(e - m[dst]); segment sum -----------------------------
__global__ void edge_seg_expsum(const int* __restrict__ ei, int E, int Nn, int H,
                                const float* __restrict__ als,
                                const float* __restrict__ ald,
                                const unsigned int* __restrict__ menc,
                                float* __restrict__ exbuf,
                                float* __restrict__ ssum) {
  int idx = blockIdx.x * blockDim.x + threadIdx.x;
  if (idx >= (E + Nn) * H) return;
  const int j = idx / H, h = idx % H;
  int s, d; edge_sd(ei, E, j, s, d);
  const float e  = leaky(als[s * H + h] + ald[d * H + h]);
  const float ex = expf(e - decOrd(menc[d * H + h]));
  exbuf[idx] = ex;
  atomicAdd(&ssum[d * H + h], ex);
}

// ---- pass 3 (layer 1, H=4,C=64): one 256-thread block per edge -------------
__global__ __launch_bounds__(256)
void edge_aggregate_h4c64(const int* __restrict__ ei, int E,
                          const float* __restrict__ exbuf,
                          const float* __restrict__ ssum,
                          const float* __restrict__ xp,
                          float* __restrict__ out) {
  const int j = blockIdx.x;                // edge id
  const int t = threadIdx.x;               // t = h*64 + c
  const int h = t >> 6;
  int s, d; edge_sd(ei, E, j, s, d);
  const float alpha = exbuf[(size_t)j * 4 + h] / (ssum[d * 4 + h] + EPSV);
  atomicAdd(&out[(size_t)d * 256 + t], alpha * xp[(size_t)s * 256 + t]);
}

// ---- pass 3 (layer 2, H=1,C=16) --------------------------------------------
__global__ void edge_aggregate_h1c16(const int* __restrict__ ei, int E, int Nn,
                                     const float* __restrict__ exbuf,
                                     const float* __restrict__ ssum,
                                     const float* __restrict__ hp,
                                     float* __restrict__ out) {
  int idx = blockIdx.x * blockDim.x + threadIdx.x;
  if (idx >= (E + Nn) * 16) return;
  const int j = idx >> 4, c = idx & 15;
  int s, d; edge_sd(ei, E, j, s, d);
  const float alpha = exbuf[j] / (ssum[d] + EPSV);
  atomicAdd(&out[(size_t)d * 16 + c], alpha * hp[(size_t)s * 16 + c]);
}

// ---- bias + ELU (in place) -------------------------------------------------
__global__ void bias_elu(float* __restrict__ h, const float* __restrict__ b,
                         size_t total, int F) {
  size_t idx = (size_t)blockIdx.x * blockDim.x + threadIdx.x;
  if (idx >= total) return;
  const float v = h[idx] + b[idx % F];
  h[idx] = v > 0.f ? v : expm1f(v);
}

// ---- bias + log_softmax over 16 classes ------------------------------------
__global__ void bias_logsoftmax16(const float* __restrict__ out2,
                                  const float* __restrict__ b,
                                  float* __restrict__ dout, int Nn) {
  int n = blockIdx.x * blockDim.x + threadIdx.x;
  if (n >= Nn) return;
  float v[16];
  float mx = -INFINITY;
#pragma unroll
  for (int c = 0; c < 16; ++c) {
    v[c] = out2[(size_t)n * 16 + c] + b[c];
    mx = fmaxf(mx, v[c]);
  }
  float s = 0.f;
#pragma unroll
  for (int c = 0; c < 16; ++c) s += expf(v[c] - mx);
  const float l = mx + logf(s);
#pragma unroll
  for (int c = 0; c < 16; ++c) dout[(size_t)n * 16 + c] = v[c] - l;
}

// ---------------------------------------------------------------------------
extern "C" void kernel_launch(void* const* d_in, const int* in_sizes, int n_in,
                              void* d_out, int out_size, void* d_ws, size_t ws_size,
                              hipStream_t stream) {
  (void)n_in; (void)out_size; (void)ws_size;
  const float* x   = (const float*)d_in[0];
  const int*   ei  = (const int*)  d_in[1];
  const float* W1  = (const float*)d_in[2];
  const float* a1s = (const float*)d_in[3];
  const float* a1d = (const float*)d_in[4];
  const float* b1  = (const float*)d_in[5];
  const float* W2  = (const float*)d_in[6];
  const float* a2s = (const float*)d_in[7];
  const float* a2d = (const float*)d_in[8];
  const float* b2  = (const float*)d_in[9];
  float* dout = (float*)d_out;

  const int F  = 256;
  const int Nn = in_sizes[0] / F;      // 50000 (multiple of 16)
  const int E  = in_sizes[1] / 2;      // 800000
  const int Et = E + Nn;               // edges incl. self-loops

  // ---- workspace layout (floats) ----
  float* ws = (float*)d_ws;
  size_t o = 0;
  float*        xp   = ws + o; o += (size_t)Nn * 256;   // layer1 projection
  float*        h1   = ws + o; o += (size_t)Nn * 256;   // layer1 aggregate / hidden
  float*        als1 = ws + o; o += (size_t)Nn * 4;
  float*        ald1 = ws + o; o += (size_t)Nn * 4;
  unsigned int* m1   = (unsigned int*)(ws + o); o += (size_t)Nn * 4;
  float*        s1   = ws + o; o += (size_t)Nn * 4;
  float*        ex1  = ws + o; o += (size_t)Et * 4;
  float*        hp   = ws + o; o += (size_t)Nn * 16;    // layer2 projection
  float*        als2 = ws + o; o += (size_t)Nn;
  float*        ald2 = ws + o; o += (size_t)Nn;
  unsigned int* m2   = (unsigned int*)(ws + o); o += (size_t)Nn;
  float*        s2   = ws + o; o += (size_t)Nn;
  float*        ex2  = ws + o; o += (size_t)Et;
  float*        out2 = ws + o; o += (size_t)Nn * 16;
  const size_t totalF = o;

  const int TB = 256;
  // zero everything (zero uint == encoded -inf for the max buffers)
  fill_zero<<<(unsigned)((totalF + TB - 1) / TB), TB, 0, stream>>>(ws, totalF);

  // ======== layer 1 ========
  gemm1_f32_wmma<<<Nn >> 4, TB, 0, stream>>>(x, W1, xp, Nn);   // 3125 blocks
  attn_logits<<<(Nn * 4 + TB - 1) / TB, TB, 0, stream>>>(xp, a1s, a1d, als1, ald1, Nn, 4, 64);
  edge_seg_max<<<(Et * 4 + TB - 1) / TB, TB, 0, stream>>>(ei, E, Nn, 4, als1, ald1, m1);
  edge_seg_expsum<<<(Et * 4 + TB - 1) / TB, TB, 0, stream>>>(ei, E, Nn, 4, als1, ald1, m1, ex1, s1);
  edge_aggregate_h4c64<<<Et, TB, 0, stream>>>(ei, E, ex1, s1, xp, h1);
  bias_elu<<<(unsigned)(((size_t)Nn * 256 + TB - 1) / TB), TB, 0, stream>>>(h1, b1, (size_t)Nn * 256, 256);

  // ======== layer 2 ========
  {
    const int tiles  = (Nn >> 4) * (16 >> 4);           // 3125
    const int blocks = (tiles + 7) / 8;
    gemm_f32_wmma<<<blocks, TB, 0, stream>>>(h1, W2, hp, Nn, 16, 256);
  }
  attn_logits<<<(Nn + TB - 1) / TB, TB, 0, stream>>>(hp, a2s, a2d, als2, ald2, Nn, 1, 16);
  edge_seg_max<<<(Et + TB - 1) / TB, TB, 0, stream>>>(ei, E, Nn, 1, als2, ald2, m2);
  edge_seg_expsum<<<(Et + TB - 1) / TB, TB, 0, stream>>>(ei, E, Nn, 1, als2, ald2, m2, ex2, s2);
  edge_aggregate_h1c16<<<(Et * 16 + TB - 1) / TB, TB, 0, stream>>>(ei, E, Nn, ex2, s2, hp, out2);

  // ======== output ========
  bias_logsoftmax16<<<(Nn + TB - 1) / TB, TB, 0, stream>>>(out2, b2, dout, Nn);
}